// CGGCN_5446018531338
// MI455X (gfx1250) — compile-verified
//
#include <hip/hip_runtime.h>
#include <hip/hip_bf16.h>

typedef __attribute__((ext_vector_type(2))) float v2f;
typedef __attribute__((ext_vector_type(8))) float v8f;

#define NN       131072      // total nodes
#define N_PER_C  4096        // nodes per subgraph
#define BATCH    32
#define E_TOT    2097152
#define D        32
#define NUM_G    14
#define CAT_STRIDE 97        // 96 + 1 pad: row*97 mod 64 hits distinct banks

#define WMMA_F32(a,b,c) \
  __builtin_amdgcn_wmma_f32_16x16x4_f32(false,(a),false,(b),(short)0,(c),false,false)

// ---------------------------------------------------------------------------
// K1: W1 = feat @ W_w + b_w ; h = feat @ self_loop_w     (16 nodes per wave)
// ---------------------------------------------------------------------------
__global__ void k1_node_linear(const float* __restrict__ feat,
                               const float* __restrict__ W_w,
                               const float* __restrict__ b_w,
                               const float* __restrict__ self_w,
                               float* __restrict__ W1,
                               float* __restrict__ h)
{
    const int lane = threadIdx.x & 31;
    const int row  = lane & 15;       // M (A) / N (B,C,D) within tile
    const int kh   = lane >> 4;       // half select
    const int wavesPerBlk = blockDim.x >> 5;
    const int gw = blockIdx.x * wavesPerBlk + (threadIdx.x >> 5);
    const int nWaves = gridDim.x * wavesPerBlk;
    const float bw0 = b_w[row], bw1 = b_w[row + 16];
    const int nGroups = NN / 16;

    for (int g = gw; g < nGroups; g += nWaves) {
        const int nb = g * 16;
        v8f aW0 = {}, aW1 = {}, aS0 = {}, aS1 = {};
        for (int k0 = 0; k0 < D; k0 += 4) {
            const int ks = k0 + 2 * kh;
            v2f a;
            const float* fp = feat + (nb + row) * D + ks;
            a.x = fp[0]; a.y = fp[1];
            v2f b0, b1, c0, c1;
            b0.x = W_w[ks * D + row];           b0.y = W_w[(ks + 1) * D + row];
            b1.x = W_w[ks * D + 16 + row];      b1.y = W_w[(ks + 1) * D + 16 + row];
            c0.x = self_w[ks * D + row];        c0.y = self_w[(ks + 1) * D + row];
            c1.x = self_w[ks * D + 16 + row];   c1.y = self_w[(ks + 1) * D + 16 + row];
            aW0 = WMMA_F32(a, b0, aW0);
            aW1 = WMMA_F32(a, b1, aW1);
            aS0 = WMMA_F32(a, c0, aS0);
            aS1 = WMMA_F32(a, c1, aS1);
        }
        for (int v = 0; v < 8; ++v) {
            const int node = nb + v + 8 * kh;
            W1[node * D + row]      = aW0[v] + bw0;
            W1[node * D + 16 + row] = aW1[v] + bw1;
            h [node * D + row]      = aS0[v];
            h [node * D + 16 + row] = aS1[v];
        }
    }
}

// ---------------------------------------------------------------------------
// K2: per-edge relation-gated MLP via WMMA, atomic segment-sum into h[dst].
//     16 edges per wave; cat tile (16x96) staged in LDS, W2 (96x32) in LDS.
// ---------------------------------------------------------------------------
__global__ void k2_edge(const float* __restrict__ W1,
                        const float* __restrict__ emb,
                        const float* __restrict__ W2,
                        const float* __restrict__ b2,
                        const float* __restrict__ norm,
                        const int* __restrict__ src,
                        const int* __restrict__ dst,
                        const int* __restrict__ etype,
                        float* __restrict__ h)
{
    __shared__ float sW2[96 * D];                    // 12 KB
    __shared__ float sCat[4][16 * CAT_STRIDE];       // ~24.8 KB (4 waves/block)
    const int tid = threadIdx.x;
    for (int i = tid; i < 96 * D; i += blockDim.x) sW2[i] = W2[i];
    __syncthreads();

    const int lane = tid & 31;
    const int wv   = tid >> 5;
    const int row  = lane & 15;
    const int kh   = lane >> 4;
    const int gw = blockIdx.x * 4 + wv;
    const int nWaves = gridDim.x * 4;
    const float b2v0 = b2[row], b2v1 = b2[row + 16];
    float* cat = sCat[wv];
    const int nGroups = E_TOT / 16;

    for (int g = gw; g < nGroups; g += nWaves) {
        const int eb = g * 16;
        // build cat = [et+w1 | et-w1 | et*w1] for 16 edges (lane = feature dim)
        for (int el = 0; el < 16; ++el) {
            const int e = eb + el;
            const int s  = src[e];
            const int et = etype[e];
            const float w1v = W1[s * D + lane];
            const float ev  = emb[et * D + lane];
            cat[el * CAT_STRIDE + lane]      = ev + w1v;
            cat[el * CAT_STRIDE + 32 + lane] = ev - w1v;
            cat[el * CAT_STRIDE + 64 + lane] = ev * w1v;
        }
        // msg_tile(16x32) = cat_tile(16x96) @ W2(96x32)
        v8f a0 = {}, a1 = {};
        for (int kc = 0; kc < 96; kc += 4) {
            const int ks = kc + 2 * kh;
            v2f a;
            a.x = cat[row * CAT_STRIDE + ks];
            a.y = cat[row * CAT_STRIDE + ks + 1];
            v2f b0, b1;
            b0.x = sW2[ks * D + row];        b0.y = sW2[(ks + 1) * D + row];
            b1.x = sW2[ks * D + 16 + row];   b1.y = sW2[(ks + 1) * D + 16 + row];
            a0 = WMMA_F32(a, b0, a0);
            a1 = WMMA_F32(a, b1, a1);
        }
        // scale by norm and scatter-add into h[dst]
        for (int v = 0; v < 8; ++v) {
            const int e  = eb + v + 8 * kh;
            const float nr = norm[e];
            const int dn = dst[e];
            atomicAdd(&h[dn * D + row],      (a0[v] + b2v0) * nr);
            atomicAdd(&h[dn * D + 16 + row], (a1[v] + b2v1) * nr);
        }
    }
}

// ---------------------------------------------------------------------------
// K3: nf = relu(h @ W_line + b_line)
// ---------------------------------------------------------------------------
__global__ void k3_line(const float* __restrict__ h,
                        const float* __restrict__ W_line,
                        const float* __restrict__ b_line,
                        float* __restrict__ nf)
{
    const int lane = threadIdx.x & 31;
    const int row  = lane & 15;
    const int kh   = lane >> 4;
    const int wavesPerBlk = blockDim.x >> 5;
    const int gw = blockIdx.x * wavesPerBlk + (threadIdx.x >> 5);
    const int nWaves = gridDim.x * wavesPerBlk;
    const float bl0 = b_line[row], bl1 = b_line[row + 16];
    const int nGroups = NN / 16;

    for (int g = gw; g < nGroups; g += nWaves) {
        const int nb = g * 16;
        v8f a0 = {}, a1 = {};
        for (int k0 = 0; k0 < D; k0 += 4) {
            const int ks = k0 + 2 * kh;
            v2f a;
            const float* hp = h + (nb + row) * D + ks;
            a.x = hp[0]; a.y = hp[1];
            v2f b0, b1;
            b0.x = W_line[ks * D + row];        b0.y = W_line[(ks + 1) * D + row];
            b1.x = W_line[ks * D + 16 + row];   b1.y = W_line[(ks + 1) * D + 16 + row];
            a0 = WMMA_F32(a, b0, a0);
            a1 = WMMA_F32(a, b1, a1);
        }
        for (int v = 0; v < 8; ++v) {
            const int node = nb + v + 8 * kh;
            nf[node * D + row]      = fmaxf(a0[v] + bl0, 0.0f);
            nf[node * D + 16 + row] = fmaxf(a1[v] + bl1, 0.0f);
        }
    }
}

// ---------------------------------------------------------------------------
// K4: per-graph (one block per graph): relation scatter, target one-hot sum,
//     4096-wide softmax path attention, zero_path fallback.
// ---------------------------------------------------------------------------
__global__ void k4_finalize(const float* __restrict__ nf,
                            const int* __restrict__ index1,
                            const int* __restrict__ index2,
                            const int* __restrict__ f2,
                            const int* __restrict__ tar_rel,
                            const float* __restrict__ zero_path,
                            float* __restrict__ out)
{
    __shared__ float sScore[N_PER_C];   // 16 KB
    __shared__ float sTgt[D];
    __shared__ float sAgg[D];
    __shared__ float sRed[256];
    __shared__ int   sAny;

    const int b   = blockIdx.x;
    const int tid = threadIdx.x;
    const int nbase = b * N_PER_C;
    float* outRel  = out + b * (NUM_G + 1) * D;
    float* outTgt  = out + BATCH * (NUM_G + 1) * D + b * D;
    float* outPath = out + BATCH * (NUM_G + 1) * D + BATCH * D + b * D;

    for (int i = tid; i < (NUM_G + 1) * D; i += blockDim.x) outRel[i] = 0.0f;
    if (tid < D) { sTgt[tid] = 0.0f; sAgg[tid] = 0.0f; }
    if (tid == 0) sAny = 0;
    __syncthreads();

    // target_rel_emd = sum over nodes with tar_rel==1
    for (int n = tid; n < N_PER_C; n += blockDim.x) {
        if (tar_rel[nbase + n] == 1) {
            const float* p = nf + (size_t)(nbase + n) * D;
            for (int d2 = 0; d2 < D; ++d2) atomicAdd(&sTgt[d2], p[d2]);
        }
    }
    __syncthreads();

    const int lane = tid & 31;      // feature dim
    const int wv   = tid >> 5;
    const int nwv  = blockDim.x >> 5;

    // relation scatter (rows 1..NUM_G; row 0 stays zero; dump slot dropped)
    for (int n = wv; n < N_PER_C; n += nwv) {
        const int node = nbase + n;
        if (index2[node] != 0) {
            const int r = f2[node] + 1;
            if (r >= 0 && r <= NUM_G)
                outRel[r * D + lane] = nf[(size_t)node * D + lane];
        }
    }

    // scores
    float lmax = -3.0e38f;
    int any = 0;
    for (int n = tid; n < N_PER_C; n += blockDim.x) {
        const int node = nbase + n;
        float s;
        if (index1[node] == 1) {
            const float* p = nf + (size_t)node * D;
            float acc = 0.0f;
            for (int d2 = 0; d2 < D; ++d2) acc += p[d2] * sTgt[d2];
            s = acc; any = 1;
        } else {
            s = -1.0e30f;
        }
        sScore[n] = s;
        lmax = fmaxf(lmax, s);
    }
    if (any) atomicOr(&sAny, 1);

    // block-reduce max
    sRed[tid] = lmax;
    __syncthreads();
    for (int off = blockDim.x >> 1; off > 0; off >>= 1) {
        if (tid < off) sRed[tid] = fmaxf(sRed[tid], sRed[tid + off]);
        __syncthreads();
    }
    const float gmax = sRed[0];
    __syncthreads();

    // block-reduce sum of exp
    float lsum = 0.0f;
    for (int n = tid; n < N_PER_C; n += blockDim.x)
        lsum += __expf(sScore[n] - gmax);
    sRed[tid] = lsum;
    __syncthreads();
    for (int off = blockDim.x >> 1; off > 0; off >>= 1) {
        if (tid < off) sRed[tid] += sRed[tid + off];
        __syncthreads();
    }
    const float inv = 1.0f / sRed[0];
    __syncthreads();

    // weighted aggregation: lane = feature dim, waves stride over nodes
    float acc = 0.0f;
    for (int n = wv; n < N_PER_C; n += nwv) {
        const float w = __expf(sScore[n] - gmax) * inv;
        acc += w * nf[(size_t)(nbase + n) * D + lane];
    }
    atomicAdd(&sAgg[lane], acc);
    __syncthreads();

    if (tid < D) {
        outTgt[tid]  = sTgt[tid];
        outPath[tid] = sAny ? sAgg[tid] : zero_path[tid];
    }
}

// ---------------------------------------------------------------------------
extern "C" void kernel_launch(void* const* d_in, const int* in_sizes, int n_in,
                              void* d_out, int out_size, void* d_ws, size_t ws_size,
                              hipStream_t stream)
{
    const float* feat   = (const float*)d_in[0];
    const float* norm   = (const float*)d_in[1];
    const float* W_w    = (const float*)d_in[2];
    const float* b_w    = (const float*)d_in[3];
    const float* W2     = (const float*)d_in[4];
    const float* b2     = (const float*)d_in[5];
    const float* emb    = (const float*)d_in[6];
    const float* selfw  = (const float*)d_in[7];
    const float* W_line = (const float*)d_in[8];
    const float* b_line = (const float*)d_in[9];
    const float* zp     = (const float*)d_in[10];
    const int* src    = (const int*)d_in[11];
    const int* dst    = (const int*)d_in[12];
    const int* etype  = (const int*)d_in[13];
    const int* index1 = (const int*)d_in[14];
    const int* index2 = (const int*)d_in[15];
    const int* f2     = (const int*)d_in[16];
    const int* tar    = (const int*)d_in[17];
    // d_in[18]=n_per, d_in[19]=num_g_rels: compile-time constants here

    float* W1 = (float*)d_ws;                 // N x 32  (16 MB)
    float* h  = W1 + (size_t)NN * D;          // N x 32  (16 MB)
    float* nf = W1;                           // reuse W1 space after K2
    float* out = (float*)d_out;

    hipLaunchKernelGGL(k1_node_linear, dim3(256),  dim3(128), 0, stream,
                       feat, W_w, b_w, selfw, W1, h);
    hipLaunchKernelGGL(k2_edge,        dim3(1024), dim3(128), 0, stream,
                       W1, emb, W2, b2, norm, src, dst, etype, h);
    hipLaunchKernelGGL(k3_line,        dim3(256),  dim3(128), 0, stream,
                       h, W_line, b_line, nf);
    hipLaunchKernelGGL(k4_finalize,    dim3(BATCH), dim3(256), 0, stream,
                       nf, index1, index2, f2, tar, zp, out);
}